// GINConvNetWithAttention_70111046140175
// MI455X (gfx1250) — compile-verified
//
#include <hip/hip_runtime.h>
#include <hip/hip_bf16.h>

// Problem constants (match reference)
#define NN 50000
#define EE 800000
#define FF 128
#define HH 128
#define LL 5
#define GG 64
#define BN_EPS 1e-5f

typedef float v2f __attribute__((ext_vector_type(2)));
typedef float v8f __attribute__((ext_vector_type(8)));

// ---------------------------------------------------------------------------
// Edge aggregation: agg[dst] += h[src].  One wave per edge, 4 feats per lane.
// ---------------------------------------------------------------------------
__global__ __launch_bounds__(256) void edge_agg_kernel(
    const float* __restrict__ h, const int* __restrict__ ei,
    float* __restrict__ agg, int e) {
  int gtid = blockIdx.x * 256 + threadIdx.x;
  int edge = gtid >> 5;
  int lane = gtid & 31;
  if (edge >= e) return;
  int src = ei[edge];
  int dst = ei[e + edge];
  const float4 v = *(const float4*)(h + (long long)src * HH + lane * 4);
  float* p = agg + (long long)dst * HH + lane * 4;
  atomicAdd(p + 0, v.x);
  atomicAdd(p + 1, v.y);
  atomicAdd(p + 2, v.z);
  atomicAdd(p + 3, v.w);
}

// ---------------------------------------------------------------------------
// GIN MLP: z = relu((h+agg) @ W1 + b1) @ W2 + b2 using V_WMMA_F32_16X16X4_F32
// One block = 16 output rows; 8 waves each own one 16-col tile.
// ---------------------------------------------------------------------------
__global__ __launch_bounds__(256) void mlp_wmma_kernel(
    const float* __restrict__ h, const float* __restrict__ agg,
    const float* __restrict__ W1, const float* __restrict__ b1,
    const float* __restrict__ W2, const float* __restrict__ b2,
    float* __restrict__ z, int n) {
  __shared__ float Xs[16][132];  // padded stride: avoid LDS bank conflicts
  __shared__ float Zs[16][132];

  int r0 = blockIdx.x * 16;
  // Cooperative load of X tile = h + agg (zeros past N)
  for (int idx = threadIdx.x; idx < 16 * HH; idx += 256) {
    int m = idx >> 7, k = idx & 127;
    int row = r0 + m;
    float v = 0.0f;
    if (row < n) v = h[(long long)row * HH + k] + agg[(long long)row * HH + k];
    Xs[m][k] = v;
  }
  __syncthreads();

  int wave = threadIdx.x >> 5;
  int lane = threadIdx.x & 31;
  int half = lane >> 4;     // 0: lanes 0-15, 1: lanes 16-31
  int mrow = lane & 15;     // row (A) / col (B,C) index within tile
  int n0 = wave * 16;       // column tile owned by this wave

  // ---- GEMM 1: X @ W1 ----
  v8f acc = {};
  for (int k0 = 0; k0 < HH; k0 += 4) {
    int kb = k0 + half * 2;  // lanes 0-15 -> K{k0,k0+1}; lanes 16-31 -> K{k0+2,k0+3}
    v2f a;
    a.x = Xs[mrow][kb];
    a.y = Xs[mrow][kb + 1];
    v2f b;
    b.x = W1[kb * HH + n0 + mrow];
    b.y = W1[(kb + 1) * HH + n0 + mrow];
    acc = __builtin_amdgcn_wmma_f32_16x16x4_f32(false, a, false, b,
                                                (short)0, acc, false, false);
  }
  // bias + ReLU -> stage z1 tile to LDS (C layout: VGPR v -> row v + 8*half)
  {
    float bias = b1[n0 + mrow];
    #pragma unroll
    for (int v = 0; v < 8; ++v) {
      float val = acc[v] + bias;
      Zs[v + half * 8][n0 + mrow] = val > 0.0f ? val : 0.0f;
    }
  }
  __syncthreads();

  // ---- GEMM 2: Z1 @ W2 ----
  v8f acc2 = {};
  for (int k0 = 0; k0 < HH; k0 += 4) {
    int kb = k0 + half * 2;
    v2f a;
    a.x = Zs[mrow][kb];
    a.y = Zs[mrow][kb + 1];
    v2f b;
    b.x = W2[kb * HH + n0 + mrow];
    b.y = W2[(kb + 1) * HH + n0 + mrow];
    acc2 = __builtin_amdgcn_wmma_f32_16x16x4_f32(false, a, false, b,
                                                 (short)0, acc2, false, false);
  }
  {
    float bias = b2[n0 + mrow];
    #pragma unroll
    for (int v = 0; v < 8; ++v) {
      int row = r0 + v + half * 8;
      if (row < n) z[(long long)row * HH + n0 + mrow] = acc2[v] + bias;
    }
  }
}

// ---------------------------------------------------------------------------
// BatchNorm statistics: stats[f] = sum, stats[128+f] = sumsq over all nodes
// ---------------------------------------------------------------------------
__global__ __launch_bounds__(128) void bn_stats_kernel(
    const float* __restrict__ z, float* __restrict__ stats, int n) {
  int f = threadIdx.x;  // 128 features
  float s = 0.0f, s2 = 0.0f;
  for (int r = blockIdx.x; r < n; r += gridDim.x) {
    float v = z[(long long)r * HH + f];
    s += v;
    s2 += v * v;
  }
  atomicAdd(&stats[f], s);
  atomicAdd(&stats[HH + f], s2);
}

// ---------------------------------------------------------------------------
// BN apply + ReLU + sigmoid attention gate + pooled (segment) sums.
// One wave per node (4 feats/lane), wave32 shuffle reduction for attention.
// ---------------------------------------------------------------------------
__global__ __launch_bounds__(256) void bn_att_pool_kernel(
    const float* __restrict__ z, const float* __restrict__ stats,
    const float* __restrict__ gamma, const float* __restrict__ beta,
    const float* __restrict__ att_w, const float* __restrict__ att_b,
    const int* __restrict__ batch, float* __restrict__ hout,
    float* __restrict__ pool, int n) {
  int gtid = blockIdx.x * 256 + threadIdx.x;
  int node = gtid >> 5;
  int lane = gtid & 31;
  if (node >= n) return;
  const float invN = 1.0f / (float)n;
  float vals[4];
  float att_partial = 0.0f;
  #pragma unroll
  for (int j = 0; j < 4; ++j) {
    int f = lane * 4 + j;
    float mu = stats[f] * invN;
    float var = stats[HH + f] * invN - mu * mu;
    float zn = (z[(long long)node * HH + f] - mu) * rsqrtf(var + BN_EPS) *
                   gamma[f] + beta[f];
    zn = zn > 0.0f ? zn : 0.0f;
    vals[j] = zn;
    att_partial += zn * att_w[f];
  }
  // wave32 all-reduce
  for (int off = 16; off > 0; off >>= 1)
    att_partial += __shfl_xor(att_partial, off, 32);
  float att = 1.0f / (1.0f + __expf(-(att_partial + att_b[0])));
  int g = batch[node];
  #pragma unroll
  for (int j = 0; j < 4; ++j) {
    int f = lane * 4 + j;
    float hv = vals[j] * att;
    hout[(long long)node * HH + f] = hv;
    atomicAdd(&pool[g * HH + f], hv);
  }
}

// ---------------------------------------------------------------------------
// Pool raw x (layer-0 head) and graph node counts
// ---------------------------------------------------------------------------
__global__ __launch_bounds__(256) void pool_x_kernel(
    const float* __restrict__ x, const int* __restrict__ batch,
    float* __restrict__ pool, int n) {
  int gtid = blockIdx.x * 256 + threadIdx.x;
  int node = gtid >> 5;
  int lane = gtid & 31;
  if (node >= n) return;
  int g = batch[node];
  const float4 v = *(const float4*)(x + (long long)node * HH + lane * 4);
  float* p = pool + g * HH + lane * 4;
  atomicAdd(p + 0, v.x);
  atomicAdd(p + 1, v.y);
  atomicAdd(p + 2, v.z);
  atomicAdd(p + 3, v.w);
}

__global__ __launch_bounds__(256) void counts_kernel(
    const int* __restrict__ batch, float* __restrict__ counts, int n) {
  int node = blockIdx.x * 256 + threadIdx.x;
  if (node < n) atomicAdd(&counts[batch[node]], 1.0f);
}

// ---------------------------------------------------------------------------
// score += (pool/denom) @ predW + predb      [G=64, H=128]
// ---------------------------------------------------------------------------
__global__ __launch_bounds__(256) void score_update_kernel(
    const float* __restrict__ pool, const float* __restrict__ counts,
    const float* __restrict__ predW, const float* __restrict__ predb,
    float* __restrict__ score) {
  int idx = blockIdx.x * 256 + threadIdx.x;  // 8192 total
  int g = idx >> 7, hcol = idx & 127;
  float inv = 1.0f / fmaxf(counts[g], 1.0f);
  float s = 0.0f;
  for (int k = 0; k < HH; ++k)
    s = fmaf(pool[g * HH + k] * inv, predW[k * HH + hcol], s);
  score[idx] += s + predb[hcol];
}

// ---------------------------------------------------------------------------
// out = relu(score @ finW + finb) @ clsW + clsb       (single block)
// ---------------------------------------------------------------------------
__global__ __launch_bounds__(256) void final_kernel(
    const float* __restrict__ score, const float* __restrict__ finW,
    const float* __restrict__ finb, const float* __restrict__ clsW,
    const float* __restrict__ clsb, float* __restrict__ out) {
  __shared__ float s1[GG][HH];
  for (int idx = threadIdx.x; idx < GG * HH; idx += 256) {
    int g = idx >> 7, hcol = idx & 127;
    float s = 0.0f;
    for (int k = 0; k < HH; ++k)
      s = fmaf(score[g * HH + k], finW[k * HH + hcol], s);
    s += finb[hcol];
    s1[g][hcol] = s > 0.0f ? s : 0.0f;
  }
  __syncthreads();
  for (int g = threadIdx.x; g < GG; g += 256) {
    float s = 0.0f;
    for (int k = 0; k < HH; ++k) s = fmaf(s1[g][k], clsW[k], s);
    out[g] = s + clsb[0];
  }
}

// ---------------------------------------------------------------------------
extern "C" void kernel_launch(void* const* d_in, const int* in_sizes, int n_in,
                              void* d_out, int out_size, void* d_ws,
                              size_t ws_size, hipStream_t stream) {
  const float* x     = (const float*)d_in[0];
  const int*   edge  = (const int*)d_in[1];
  const int*   batch = (const int*)d_in[2];
  const float* W1    = (const float*)d_in[3];
  const float* b1    = (const float*)d_in[4];
  const float* W2    = (const float*)d_in[5];
  const float* b2    = (const float*)d_in[6];
  const float* gamma = (const float*)d_in[7];
  const float* beta  = (const float*)d_in[8];
  const float* att_w = (const float*)d_in[9];
  const float* att_b = (const float*)d_in[10];
  const float* predW = (const float*)d_in[11];
  const float* predb = (const float*)d_in[12];
  const float* finW  = (const float*)d_in[13];
  const float* finb  = (const float*)d_in[14];
  const float* clsW  = (const float*)d_in[15];
  const float* clsb  = (const float*)d_in[16];
  float* out = (float*)d_out;

  const long long NH = (long long)NN * HH;  // 6.4M floats
  float* h      = (float*)d_ws;
  float* agg    = h + NH;
  float* z      = agg + NH;
  float* stats  = z + NH;        // 256
  float* counts = stats + 256;   // 64
  float* pool   = counts + GG;   // 8192
  float* score  = pool + GG * HH;  // 8192

  // init per-call state (deterministic; safe under graph capture)
  hipMemcpyAsync(h, x, NH * sizeof(float), hipMemcpyDeviceToDevice, stream);
  hipMemsetAsync(counts, 0, GG * sizeof(float), stream);
  hipMemsetAsync(pool, 0, GG * HH * sizeof(float), stream);
  hipMemsetAsync(score, 0, GG * HH * sizeof(float), stream);

  counts_kernel<<<(NN + 255) / 256, 256, 0, stream>>>(batch, counts, NN);
  pool_x_kernel<<<((long long)NN * 32 + 255) / 256, 256, 0, stream>>>(
      x, batch, pool, NN);
  score_update_kernel<<<(GG * HH) / 256, 256, 0, stream>>>(
      pool, counts, predW, predb, score);

  const int mlp_blocks = (NN + 15) / 16;  // 3125
  for (int i = 0; i < LL; ++i) {
    hipMemsetAsync(agg, 0, NH * sizeof(float), stream);
    edge_agg_kernel<<<((long long)EE * 32 + 255) / 256, 256, 0, stream>>>(
        h, edge, agg, EE);
    mlp_wmma_kernel<<<mlp_blocks, 256, 0, stream>>>(
        h, agg, W1 + (long long)i * FF * HH, b1 + i * HH,
        W2 + (long long)i * HH * HH, b2 + i * HH, z, NN);
    hipMemsetAsync(stats, 0, 2 * HH * sizeof(float), stream);
    bn_stats_kernel<<<256, 128, 0, stream>>>(z, stats, NN);
    hipMemsetAsync(pool, 0, GG * HH * sizeof(float), stream);
    bn_att_pool_kernel<<<((long long)NN * 32 + 255) / 256, 256, 0, stream>>>(
        z, stats, gamma + i * HH, beta + i * HH, att_w + i * HH, att_b + i,
        batch, h, pool, NN);
    score_update_kernel<<<(GG * HH) / 256, 256, 0, stream>>>(
        pool, counts, predW + (long long)(i + 1) * HH * HH,
        predb + (i + 1) * HH, score);
  }

  final_kernel<<<1, 256, 0, stream>>>(score, finW, finb, clsW, clsb, out);
}